// ConvectionR2_10101763080235
// MI455X (gfx1250) — compile-verified
//
#include <hip/hip_runtime.h>
#include <cstdint>
#include <cstddef>

// Problem constants (match reference)
#define Bdim 8
#define Cdim 128
#define Hdim 256
#define Wdim 256

#define TROWS 32                                  // output rows per workgroup
#define RPT 8                                     // output rows per thread
#define CPT 4                                     // output cols per thread
#define NTHREADS 256                              // 8 wave32 waves
#define SROWS (TROWS + 1)                         // 33 staged rows
#define TILE_ELEMS (SROWS * (Wdim + 1))           // 8481 floats (~33.9 KB)

typedef uint32_t u32x4 __attribute__((ext_vector_type(4)));
typedef int      i32x4 __attribute__((ext_vector_type(4)));
typedef int      i32x8 __attribute__((ext_vector_type(8)));
typedef float    f32x4 __attribute__((ext_vector_type(4)));

__global__ __launch_bounds__(NTHREADS)
void ConvectionR2_kernel(const float* __restrict__ in,
                         const float* __restrict__ c,
                         float* __restrict__ out)
{
    // Packed staged rect: yc rows x xc cols (row stride == xc), yc<=33, xc<=257.
    __shared__ float tile[TILE_ELEMS];

    const int tilesPerPlane = Hdim / TROWS;              // 8
    const int plane = blockIdx.x / tilesPerPlane;        // b*C + ch
    const int h0    = (blockIdx.x % tilesPerPlane) * TROWS;
    const int ch    = plane & (Cdim - 1);
    const int tid   = threadIdx.x;

    // Per-channel constant shift: floor(h - cy) == h + floor(-cy) for integer h.
    const float fy = -c[2 * ch + 0];
    const float fx = -c[2 * ch + 1];
    const float fdy = floorf(fy), fdx = floorf(fx);
    const int   dy = (int)fdy,    dx = (int)fdx;
    const float wy = fy - fdy,    wx = fx - fdx;

    const float* inp = in + (size_t)plane * (size_t)(Hdim * Wdim);

    // Needed window: rows [y_start, y_start+33) x cols [x_start, x_start+257),
    // intersected with the plane [0,H) x [0,W).  (All wave-uniform -> SGPRs.)
    const int y_start = h0 + dy;
    const int x_start = dx;
    int y0c = y_start < 0 ? 0 : (y_start > Hdim - 1 ? Hdim - 1 : y_start);
    int x0c = x_start < 0 ? 0 : (x_start > Wdim - 1 ? Wdim - 1 : x_start);
    int y1c = y_start + TROWS + 1; if (y1c > Hdim) y1c = Hdim;
    int x1c = x_start + Wdim + 1;  if (x1c > Wdim) x1c = Wdim;
    int yc = y1c - y0c; if (yc < 0) yc = 0;              // tile_dim1 (0 => TDM NOP)
    int xc = x1c - x0c; if (xc < 0) xc = 0;              // tile_dim0 (0 => TDM NOP)

    const uint32_t lds_base = (uint32_t)(uintptr_t)(&tile[0]);   // low 32 bits = DS byte addr
    const uint64_t ga = (uint64_t)(uintptr_t)(inp + (size_t)y0c * Wdim + x0c);

#if defined(__has_builtin) && __has_builtin(__builtin_amdgcn_tensor_load_to_lds)
    // ---- Tensor Data Mover: one DMA of the whole staged rect, issued by wave 0 ----
    if (tid < 32) {
        // D# group 0: count=1 | lds_addr | global_addr[56:0] | type=2 (bits 127:126)
        u32x4 g0 = { 1u,
                     lds_base,
                     (uint32_t)ga,
                     ((uint32_t)(ga >> 32) & 0x01FFFFFFu) | 0x80000000u };
        // D# group 1: data_size=4B; tensor_dim0/1 = tile_dim0/1 = (xc,yc) so no TDM-side
        // clipping; tensor_dim0_stride = W; rows packed in LDS (stride xc).
        i32x8 g1 = { 0x00020000,
                     (xc & 0xFFFF) << 16,     // bits 63:48  = tensor_dim0[15:0]
                     (yc & 0xFFFF) << 16,     // bits 95:80  = tensor_dim1[15:0]
                     (xc & 0xFFFF) << 16,     // bits 127:112 = tile_dim0
                     (yc & 0xFFFF),           // bits 143:128 = tile_dim1 (tile_dim2=0)
                     Wdim,                    // bits 191:160 = tensor_dim0_stride[31:0]
                     0, 0 };
        i32x4 gz4 = { 0, 0, 0, 0 };
        i32x8 gz8 = { 0, 0, 0, 0, 0, 0, 0, 0 };
        __builtin_amdgcn_tensor_load_to_lds(g0, g1, gz4, gz4, gz8, 0);
        __builtin_amdgcn_s_wait_tensorcnt(0);
    }
#else
    // ---- Fallback: per-lane async global->LDS staging of the same packed rect ----
    for (int r = 0; r < yc; ++r) {
        const float* rowp = inp + (size_t)(y0c + r) * Wdim + x0c;
        for (int j = tid; j < xc; j += NTHREADS) {
            uint32_t dst = lds_base + (uint32_t)(r * xc + j) * 4u;
            asm volatile("global_load_async_to_lds_b32 %0, %1, off"
                         :: "v"(dst), "v"(rowp + j) : "memory");
        }
    }
    asm volatile("s_wait_asynccnt 0" ::: "memory");
#endif
    __syncthreads();

    // ---- Compute: 4 cols x 8 rows per thread ----
    const int cg = tid & 63;                 // column group -> cols [4cg, 4cg+3]
    const int rg = tid >> 6;                 // row group    -> rows [8rg, 8rg+7]
    const int colbase = cg * CPT;
    const int jl = colbase + x_start - x0c;  // staged col of tap 0 (exact when valid)
    const float cw0 = 1.0f - wx, cw1 = wx;
    const float rw0 = 1.0f - wy, rw1 = wy;
    const int xcm1 = (xc > 0) ? (xc - 1) : 0;

    // Per-tap column validity + clamped staged column (hoisted out of row loop).
    bool vx[CPT + 1];
    int  colc[CPT + 1];
    #pragma unroll
    for (int k = 0; k <= CPT; ++k) {
        int xg = colbase + k + x_start;                      // global tap column
        vx[k] = (unsigned)xg < (unsigned)Wdim;
        int cc = xg - x0c;
        if (cc < 0) cc = 0; if (cc > xcm1) cc = xcm1;        // exact when vx[k]
        colc[k] = cc;
    }
    const bool vxall = vx[0] && vx[1] && vx[2] && vx[3] && vx[4];

    // Horizontal lerp of the staged row feeding output row r (0..RPT).
    // Selects (not 0-weights) kill OOB taps so unwritten-LDS garbage/NaN can
    // never propagate; all indices provably in [0, TILE_ELEMS).
    auto rowv = [&](int r, bool fast) -> f32x4 {
        int y0 = y_start + rg * RPT + r;                     // tap row
        int rloc = y0 - y0c;
        if (rloc < 0) rloc = 0; if (rloc > SROWS - 1) rloc = SROWS - 1;
        const bool vy = (unsigned)y0 < (unsigned)Hdim;
        const int base = rloc * xc;
        float t[CPT + 1];
        if (fast) {
            const int b = base + jl;                         // jl in [0, xc-5] here
            #pragma unroll
            for (int k = 0; k <= CPT; ++k) t[k] = tile[b + k];
        } else {
            #pragma unroll
            for (int k = 0; k <= CPT; ++k) t[k] = vx[k] ? tile[base + colc[k]] : 0.0f;
        }
        f32x4 rv;
        #pragma unroll
        for (int k = 0; k < CPT; ++k) {
            float v = cw0 * t[k] + cw1 * t[k + 1];
            rv[k] = vy ? v : 0.0f;
        }
        return rv;
    };

    float* outp = out + ((size_t)plane * Hdim + h0 + rg * RPT) * (size_t)Wdim + colbase;

    if (__all(vxall)) {                                      // wave-uniform fast path
        f32x4 prev = rowv(0, true);
        #pragma unroll
        for (int rr = 0; rr < RPT; ++rr) {
            f32x4 next = rowv(rr + 1, true);
            f32x4 o;
            #pragma unroll
            for (int k = 0; k < CPT; ++k) o[k] = rw0 * prev[k] + rw1 * next[k];
            __builtin_nontemporal_store(o, (f32x4*)(outp + (size_t)rr * Wdim));
            prev = next;
        }
    } else {                                                 // edge waves
        f32x4 prev = rowv(0, false);
        #pragma unroll
        for (int rr = 0; rr < RPT; ++rr) {
            f32x4 next = rowv(rr + 1, false);
            f32x4 o;
            #pragma unroll
            for (int k = 0; k < CPT; ++k) o[k] = rw0 * prev[k] + rw1 * next[k];
            __builtin_nontemporal_store(o, (f32x4*)(outp + (size_t)rr * Wdim));
            prev = next;
        }
    }
}

extern "C" void kernel_launch(void* const* d_in, const int* in_sizes, int n_in,
                              void* d_out, int out_size, void* d_ws, size_t ws_size,
                              hipStream_t stream) {
    (void)in_sizes; (void)n_in; (void)out_size; (void)d_ws; (void)ws_size;
    const float* in  = (const float*)d_in[0];   // [B,C,H,W] fp32
    const float* c   = (const float*)d_in[1];   // [C,2] fp32
    float*       out = (float*)d_out;           // [B,C,H,W] fp32

    dim3 grid(Bdim * Cdim * (Hdim / TROWS));    // 8192 workgroups
    dim3 block(NTHREADS);
    ConvectionR2_kernel<<<grid, block, 0, stream>>>(in, c, out);
}